// GCN_33784212750512
// MI455X (gfx1250) — compile-verified
//
#include <hip/hip_runtime.h>
#include <hip/hip_bf16.h>

typedef __attribute__((ext_vector_type(2))) float v2f;
typedef __attribute__((ext_vector_type(8))) float v8f;

#define F 128          // feature width (F_IN == HID == 128)

// ---------------------------------------------------------------------------
// Degree / normalization
// ---------------------------------------------------------------------------
__global__ void init_one_kernel(float* __restrict__ deg, int n) {
    int i = blockIdx.x * blockDim.x + threadIdx.x;
    if (i < n) deg[i] = 1.0f;                     // self-loop contribution
}

__global__ void count_dst_kernel(const int* __restrict__ ei, float* __restrict__ deg, int e) {
    int i = blockIdx.x * blockDim.x + threadIdx.x;
    if (i < e) atomicAdd(&deg[ei[e + i]], 1.0f);  // dst row of edge_index
}

__global__ void rsqrt_inplace_kernel(float* __restrict__ deg, int n) {
    int i = blockIdx.x * blockDim.x + threadIdx.x;
    if (i < n) deg[i] = rsqrtf(deg[i]);           // deg >= 1 always
}

// ---------------------------------------------------------------------------
// Dense GEMM: out[N x 128] = A[N x 128] @ W[128 x 128], exact fp32 via
// V_WMMA_F32_16X16X4_F32.  Block = 256 threads = 8 waves; block handles a
// 16-row strip, wave w computes the 16x16 tile at columns [16w, 16w+16).
// ---------------------------------------------------------------------------
__global__ __launch_bounds__(256) void gemm_f32_wmma_kernel(
    const float* __restrict__ A, const float* __restrict__ W,
    float* __restrict__ out, int nrows)
{
    const int lane = threadIdx.x & 31;
    const int wave = threadIdx.x >> 5;            // 0..7  -> column tile
    const int row0 = blockIdx.x * 16;
    if (row0 >= nrows) return;

    const int mn   = lane & 15;                   // A row in tile / B,D column
    const int kh   = (lane >> 4) * 2;             // K sub-offset for half-wave
    const int col  = wave * 16 + mn;              // global output column

    const float* arow = A + (size_t)(row0 + mn) * F;

    v8f acc0 = {};
    v8f acc1 = {};
#pragma unroll
    for (int k = 0; k < F; k += 8) {
        v2f a0, b0, a1, b1;
        a0.x = arow[k + kh];
        a0.y = arow[k + kh + 1];
        b0.x = W[(size_t)(k + kh) * F + col];
        b0.y = W[(size_t)(k + kh + 1) * F + col];
        acc0 = __builtin_amdgcn_wmma_f32_16x16x4_f32(false, a0, false, b0,
                                                     (short)0, acc0, false, false);
        a1.x = arow[k + 4 + kh];
        a1.y = arow[k + 4 + kh + 1];
        b1.x = W[(size_t)(k + 4 + kh) * F + col];
        b1.y = W[(size_t)(k + 4 + kh + 1) * F + col];
        acc1 = __builtin_amdgcn_wmma_f32_16x16x4_f32(false, a1, false, b1,
                                                     (short)0, acc1, false, false);
    }
    v8f acc = acc0 + acc1;

    const int mstore = 8 * (lane >> 4);           // D: M = r + 8*(lane>=16)
#pragma unroll
    for (int r = 0; r < 8; ++r)
        out[(size_t)(row0 + mstore + r) * F + col] = acc[r];
}

// ---------------------------------------------------------------------------
// agg[n][:] = dinv[n]^2 * tmp[n][:]   (self-loop seed; initializes agg)
// ---------------------------------------------------------------------------
__global__ __launch_bounds__(256) void seed_self_kernel(
    const float* __restrict__ tmp, const float* __restrict__ dinv,
    float* __restrict__ agg, int n)
{
    int t = blockIdx.x * blockDim.x + threadIdx.x;
    int node = t >> 5;
    if (node >= n) return;
    int lane = t & 31;
    float s = dinv[node];
    s = s * s;
    float4 v = ((const float4*)(tmp + (size_t)node * F))[lane];
    ((float4*)(agg + (size_t)node * F))[lane] =
        make_float4(v.x * s, v.y * s, v.z * s, v.w * s);
}

// ---------------------------------------------------------------------------
// Edge scatter: one wave per edge. agg[dst] += dinv[s]*dinv[d] * tmp[src]
// ---------------------------------------------------------------------------
__global__ __launch_bounds__(256) void scatter_feat_kernel(
    const int* __restrict__ ei, const float* __restrict__ tmp,
    const float* __restrict__ dinv, float* __restrict__ agg, int nedges)
{
    int e = (blockIdx.x * blockDim.x + threadIdx.x) >> 5;
    if (e >= nedges) return;
    int lane = threadIdx.x & 31;
    int s = ei[e];
    int d = ei[nedges + e];
    float nrm = dinv[s] * dinv[d];
    float4 v = ((const float4*)(tmp + (size_t)s * F))[lane];
    float* dst = agg + (size_t)d * F + lane * 4;
    atomicAdd(dst + 0, v.x * nrm);
    atomicAdd(dst + 1, v.y * nrm);
    atomicAdd(dst + 2, v.z * nrm);
    atomicAdd(dst + 3, v.w * nrm);
}

// ---------------------------------------------------------------------------
// In-place h = relu(agg + b)
// ---------------------------------------------------------------------------
__global__ __launch_bounds__(256) void bias_relu_kernel(
    float* __restrict__ agg, const float* __restrict__ b, int total)
{
    int i = blockIdx.x * blockDim.x + threadIdx.x;
    if (i >= total) return;
    float v = agg[i] + b[i & (F - 1)];
    agg[i] = v > 0.0f ? v : 0.0f;
}

// ---------------------------------------------------------------------------
// Layer 3: s3[n] = dot(h[n], W3), one wave per node, shfl_xor reduction
// ---------------------------------------------------------------------------
__global__ __launch_bounds__(256) void dot128_kernel(
    const float* __restrict__ h, const float* __restrict__ W3,
    float* __restrict__ s3, int n)
{
    int node = (blockIdx.x * blockDim.x + threadIdx.x) >> 5;
    if (node >= n) return;
    int lane = threadIdx.x & 31;
    float4 hv = ((const float4*)(h + (size_t)node * F))[lane];
    float4 wv = ((const float4*)W3)[lane];
    float v = hv.x * wv.x + hv.y * wv.y + hv.z * wv.z + hv.w * wv.w;
#pragma unroll
    for (int off = 16; off > 0; off >>= 1) v += __shfl_xor(v, off, 32);
    if (lane == 0) s3[node] = v;
}

__global__ void seed3_kernel(const float* __restrict__ s3,
                             const float* __restrict__ dinv,
                             float* __restrict__ a3, int n)
{
    int i = blockIdx.x * blockDim.x + threadIdx.x;
    if (i >= n) return;
    float s = dinv[i];
    a3[i] = s * s * s3[i];
}

__global__ void scatter3_kernel(const int* __restrict__ ei,
                                const float* __restrict__ s3,
                                const float* __restrict__ dinv,
                                float* __restrict__ a3, int nedges)
{
    int i = blockIdx.x * blockDim.x + threadIdx.x;
    if (i >= nedges) return;
    int s = ei[i];
    int d = ei[nedges + i];
    atomicAdd(&a3[d], dinv[s] * dinv[d] * s3[s]);
}

__global__ void final_out_kernel(const float* __restrict__ a3,
                                 const float* __restrict__ b3,
                                 const float* __restrict__ x,
                                 float* __restrict__ out, int n)
{
    int i = blockIdx.x * blockDim.x + threadIdx.x;
    if (i >= n) return;
    float v = a3[i] + b3[0];
    v = v > 0.0f ? v : 0.0f;
    out[i] = v + x[(size_t)i * F + (F - 1)];     // + x[:, -1]
}

// ---------------------------------------------------------------------------
extern "C" void kernel_launch(void* const* d_in, const int* in_sizes, int n_in,
                              void* d_out, int out_size, void* d_ws, size_t ws_size,
                              hipStream_t stream)
{
    const float* x  = (const float*)d_in[0];
    const int*   ei = (const int*)d_in[1];
    // d_in[2] = batch (unused: B*NUM_NODES layout is contiguous)
    const float* W1 = (const float*)d_in[3];
    const float* b1 = (const float*)d_in[4];
    const float* W2 = (const float*)d_in[5];
    const float* b2 = (const float*)d_in[6];
    const float* W3 = (const float*)d_in[7];
    const float* b3 = (const float*)d_in[8];
    float* out = (float*)d_out;

    const int N = in_sizes[0] / F;       // 100000
    const int E = in_sizes[1] / 2;       // 1600000

    const size_t featBytes = (size_t)N * F * sizeof(float);
    const size_t need = 2 * featBytes + 3 * (size_t)N * sizeof(float);
    if (ws_size < need) return;

    char* ws = (char*)d_ws;
    float* bufA = (float*)ws;                        // N x 128 (GEMM out)
    float* bufB = (float*)(ws + featBytes);          // N x 128 (agg / h)
    float* dinv = (float*)(ws + 2 * featBytes);      // N  (deg -> dinv in place)
    float* s3   = dinv + N;                          // N
    float* a3   = s3 + N;                            // N

    const int T = 256;
    const int nBlkN   = (N + T - 1) / T;
    const int nBlkE   = (E + T - 1) / T;
    const int nBlkNW  = (N * 32 + T - 1) / T;        // wave-per-node
    const int nBlkEW  = (int)(((size_t)E * 32 + T - 1) / T);  // wave-per-edge
    const int nBlkNF  = (N * F + T - 1) / T;
    const int nBlkGemm = (N + 15) / 16;

    // --- normalization: deg -> dinv ---
    init_one_kernel<<<nBlkN, T, 0, stream>>>(dinv, N);
    count_dst_kernel<<<nBlkE, T, 0, stream>>>(ei, dinv, E);
    rsqrt_inplace_kernel<<<nBlkN, T, 0, stream>>>(dinv, N);

    // --- layer 1 ---
    gemm_f32_wmma_kernel<<<nBlkGemm, T, 0, stream>>>(x, W1, bufA, N);
    seed_self_kernel<<<nBlkNW, T, 0, stream>>>(bufA, dinv, bufB, N);
    scatter_feat_kernel<<<nBlkEW, T, 0, stream>>>(ei, bufA, dinv, bufB, E);
    bias_relu_kernel<<<nBlkNF, T, 0, stream>>>(bufB, b1, N * F);

    // --- layer 2 ---
    gemm_f32_wmma_kernel<<<nBlkGemm, T, 0, stream>>>(bufB, W2, bufA, N);
    seed_self_kernel<<<nBlkNW, T, 0, stream>>>(bufA, dinv, bufB, N);
    scatter_feat_kernel<<<nBlkEW, T, 0, stream>>>(ei, bufA, dinv, bufB, E);
    bias_relu_kernel<<<nBlkNF, T, 0, stream>>>(bufB, b2, N * F);

    // --- layer 3 (HID -> 1) + residual ---
    dot128_kernel<<<nBlkNW, T, 0, stream>>>(bufB, W3, s3, N);
    seed3_kernel<<<nBlkN, T, 0, stream>>>(s3, dinv, a3, N);
    scatter3_kernel<<<nBlkE, T, 0, stream>>>(ei, s3, dinv, a3, E);
    final_out_kernel<<<nBlkN, T, 0, stream>>>(a3, b3, x, out, N);
}